// MultiHeadSelfAttention_19868518711370
// MI455X (gfx1250) — compile-verified
//
#include <hip/hip_runtime.h>
#include <hip/hip_bf16.h>

// Problem constants (from reference)
#define BB   8
#define SS   2048
#define FF   256
#define HH   8
#define DD   64
#define PROJ 512   // HH * DD
#define HP   16    // heads padded to WMMA tile width

typedef __attribute__((ext_vector_type(2))) float v2f;
typedef __attribute__((ext_vector_type(8))) float v8f;

// ---------------------------------------------------------------------------
// Kernel 1: kq16[b,h,f] = (1/sqrt(D)) * sum_d Wk[f, h*D+d] * q_last[b, h*D+d]
//           where q_last[b,p] = sum_f x[b,S-1,f] * Wq[f,p]
// Head rows 8..15 are written as zeros (WMMA padding).
// grid(B), block(256)
// ---------------------------------------------------------------------------
__global__ void mhsa_kq_kernel(const float* __restrict__ x,
                               const float* __restrict__ Wq,
                               const float* __restrict__ Wk,
                               float* __restrict__ kq16) {
    __shared__ float xs[FF];
    __shared__ float qs[PROJ];
    const int b = blockIdx.x;
    const int t = threadIdx.x;

    xs[t] = x[((size_t)b * SS + (SS - 1)) * FF + t];
    __syncthreads();

    for (int p = t; p < PROJ; p += 256) {
        float acc = 0.f;
        for (int f = 0; f < FF; ++f) acc += xs[f] * Wq[(size_t)f * PROJ + p];
        qs[p] = acc;
    }
    __syncthreads();

    const float scale = 0.125f;  // 1/sqrt(64)
    const float* wkRow = Wk + (size_t)t * PROJ;  // t == f
    for (int h = 0; h < HH; ++h) {
        float acc = 0.f;
        #pragma unroll 8
        for (int d = 0; d < DD; ++d) acc += wkRow[h * DD + d] * qs[h * DD + d];
        kq16[((size_t)b * HP + h) * FF + t] = acc * scale;
    }
    for (int h = HH; h < HP; ++h)            // zero padding rows
        kq16[((size_t)b * HP + h) * FF + t] = 0.f;
}

// ---------------------------------------------------------------------------
// Kernel 2: scores via WMMA f32 16x16x4.
// Each wave owns 4 s-tiles (64 rows) x 16 head-columns; one shared B fragment
// per k-step feeds 4 independent accumulator chains.
// Stores the full padded w16[b, head16, s] (pad columns are exactly zero).
// grid(B, S/256), block(128) = 4 waves
// ---------------------------------------------------------------------------
__global__ void mhsa_score_kernel(const float* __restrict__ x,
                                  const float* __restrict__ kq16,
                                  float* __restrict__ w16) {
    const int b     = blockIdx.x;
    const int wave  = threadIdx.x >> 5;
    const int lane  = threadIdx.x & 31;
    const int sWave = blockIdx.y * 256 + wave * 64;

    const int mn    = lane & 15;          // A: row s-offset, B: column (head)
    const int khalf = (lane >> 4) << 1;   // 0 or 2

    const float* xr0   = x + ((size_t)b * SS + sWave + mn) * FF;
    const float* kqrow = kq16 + ((size_t)b * HP + mn) * FF;

    v8f acc0 = {0.f,0.f,0.f,0.f,0.f,0.f,0.f,0.f};
    v8f acc1 = acc0, acc2 = acc0, acc3 = acc0;

    for (int k0 = 0; k0 < FF; k0 += 4) {
        v2f bf;
        bf[0] = kqrow[k0 + khalf];
        bf[1] = kqrow[k0 + khalf + 1];
        v2f a0, a1, a2, a3;
        a0[0] = xr0[k0 + khalf];                 a0[1] = xr0[k0 + khalf + 1];
        a1[0] = xr0[16 * FF + k0 + khalf];       a1[1] = xr0[16 * FF + k0 + khalf + 1];
        a2[0] = xr0[32 * FF + k0 + khalf];       a2[1] = xr0[32 * FF + k0 + khalf + 1];
        a3[0] = xr0[48 * FF + k0 + khalf];       a3[1] = xr0[48 * FF + k0 + khalf + 1];
        acc0 = __builtin_amdgcn_wmma_f32_16x16x4_f32(false, a0, false, bf, (short)0, acc0, false, false);
        acc1 = __builtin_amdgcn_wmma_f32_16x16x4_f32(false, a1, false, bf, (short)0, acc1, false, false);
        acc2 = __builtin_amdgcn_wmma_f32_16x16x4_f32(false, a2, false, bf, (short)0, acc2, false, false);
        acc3 = __builtin_amdgcn_wmma_f32_16x16x4_f32(false, a3, false, bf, (short)0, acc3, false, false);
    }

    // D: lane l, vgpr r -> [s = tile + r + 8*(l>>4), head = l&15]
    const int mhi = (lane >> 4) << 3;
    float* out = w16 + ((size_t)b * HP + mn) * SS + sWave + mhi;
    #pragma unroll
    for (int r = 0; r < 8; ++r) out[r]      = acc0[r];
    #pragma unroll
    for (int r = 0; r < 8; ++r) out[16 + r] = acc1[r];
    #pragma unroll
    for (int r = 0; r < 8; ++r) out[32 + r] = acc2[r];
    #pragma unroll
    for (int r = 0; r < 8; ++r) out[48 + r] = acc3[r];
}

// ---------------------------------------------------------------------------
// Kernel 3: in-place softmax over S for the 8 real head rows of each batch.
// grid(B*H), block(256). Pad rows (8..15) stay zero.
// ---------------------------------------------------------------------------
__global__ void mhsa_softmax_kernel(float* __restrict__ w16) {
    const int t    = threadIdx.x;
    const int lane = t & 31;
    const int wid  = t >> 5;
    const int b    = blockIdx.x >> 3;
    const int h    = blockIdx.x & 7;
    float* row = w16 + ((size_t)b * HP + h) * SS;
    __shared__ float sred[8];

    float lmax = -3.0e38f;
    for (int i = t; i < SS; i += 256) lmax = fmaxf(lmax, row[i]);
    #pragma unroll
    for (int off = 16; off > 0; off >>= 1) lmax = fmaxf(lmax, __shfl_xor(lmax, off));
    if (lane == 0) sred[wid] = lmax;
    __syncthreads();
    float m = sred[0];
    #pragma unroll
    for (int i = 1; i < 8; ++i) m = fmaxf(m, sred[i]);
    __syncthreads();  // everyone has read sred; safe to reuse

    float lsum = 0.f;
    for (int i = t; i < SS; i += 256) {
        float e = __expf(row[i] - m);
        row[i] = e;
        lsum += e;
    }
    #pragma unroll
    for (int off = 16; off > 0; off >>= 1) lsum += __shfl_xor(lsum, off);
    if (lane == 0) sred[wid] = lsum;
    __syncthreads();
    float z = 0.f;
    #pragma unroll
    for (int i = 0; i < 8; ++i) z += sred[i];
    const float inv = 1.0f / z;
    for (int i = t; i < SS; i += 256) row[i] *= inv;
}

// ---------------------------------------------------------------------------
// Kernel 4: ctx[b,h,f] = sum_s w16[b,h,s] * x[b,s,f]  via WMMA f32 16x16x4.
// A[m=head16, k=s] (pad rows already zero), B[k=s, n=f]. Each wave covers a
// 512-wide s-slice and 4 f-tiles sharing one A fragment per k-step; partial
// tiles reduced through LDS. grid(B, F/64), block(128)
// ---------------------------------------------------------------------------
__global__ void mhsa_ctx_kernel(const float* __restrict__ x,
                                const float* __restrict__ w16,
                                float* __restrict__ ctx) {
    const int b    = blockIdx.x;
    const int f0   = blockIdx.y * 64;
    const int wave = threadIdx.x >> 5;
    const int lane = threadIdx.x & 31;

    const int mn    = lane & 15;          // A: head row, B: f column
    const int khalf = (lane >> 4) << 1;   // 0 or 2

    const float* wr = w16 + ((size_t)b * HP + mn) * SS;

    v8f acc0 = {0.f,0.f,0.f,0.f,0.f,0.f,0.f,0.f};
    v8f acc1 = acc0, acc2 = acc0, acc3 = acc0;

    const int sEnd = wave * 512 + 512;
    for (int s0 = wave * 512; s0 < sEnd; s0 += 4) {
        v2f a;
        a[0] = wr[s0 + khalf];
        a[1] = wr[s0 + khalf + 1];
        const float* xp = x + ((size_t)b * SS + s0 + khalf) * FF + f0 + mn;
        v2f b0, b1, b2, b3;
        b0[0] = xp[0];      b0[1] = xp[FF];
        b1[0] = xp[16];     b1[1] = xp[FF + 16];
        b2[0] = xp[32];     b2[1] = xp[FF + 32];
        b3[0] = xp[48];     b3[1] = xp[FF + 48];
        acc0 = __builtin_amdgcn_wmma_f32_16x16x4_f32(false, a, false, b0, (short)0, acc0, false, false);
        acc1 = __builtin_amdgcn_wmma_f32_16x16x4_f32(false, a, false, b1, (short)0, acc1, false, false);
        acc2 = __builtin_amdgcn_wmma_f32_16x16x4_f32(false, a, false, b2, (short)0, acc2, false, false);
        acc3 = __builtin_amdgcn_wmma_f32_16x16x4_f32(false, a, false, b3, (short)0, acc3, false, false);
    }

    __shared__ float tile[4][4][256];     // [wave][f-tile][16x16]
    const int mhi = (lane >> 4) << 3;
    #pragma unroll
    for (int r = 0; r < 8; ++r) tile[wave][0][(mhi + r) * 16 + mn] = acc0[r];
    #pragma unroll
    for (int r = 0; r < 8; ++r) tile[wave][1][(mhi + r) * 16 + mn] = acc1[r];
    #pragma unroll
    for (int r = 0; r < 8; ++r) tile[wave][2][(mhi + r) * 16 + mn] = acc2[r];
    #pragma unroll
    for (int r = 0; r < 8; ++r) tile[wave][3][(mhi + r) * 16 + mn] = acc3[r];
    __syncthreads();

    // Useful outputs per f-tile: heads 0..7 x 16 columns = 128, one per thread
    const int m = threadIdx.x >> 4;  // head 0..7
    const int n = threadIdx.x & 15;  // f offset
    #pragma unroll
    for (int ft = 0; ft < 4; ++ft) {
        const float v = tile[0][ft][m * 16 + n] + tile[1][ft][m * 16 + n] +
                        tile[2][ft][m * 16 + n] + tile[3][ft][m * 16 + n];
        ctx[((size_t)b * HH + m) * FF + f0 + ft * 16 + n] = v;
    }
}

// ---------------------------------------------------------------------------
// Kernel 5: attn[b,p] = sum_f ctx[b, p/D, f] * Wv[f,p]
//           out[b,f]  = sum_p attn[b,p] * Wo[p,f] + bo[f]
// grid(B), block(256)
// ---------------------------------------------------------------------------
__global__ void mhsa_out_kernel(const float* __restrict__ ctx,
                                const float* __restrict__ Wv,
                                const float* __restrict__ Wo,
                                const float* __restrict__ bo,
                                float* __restrict__ out) {
    __shared__ float ctxs[HH * FF];   // 8 KB
    __shared__ float attns[PROJ];
    const int b = blockIdx.x;
    const int t = threadIdx.x;

    for (int i = t; i < HH * FF; i += 256) ctxs[i] = ctx[(size_t)b * HH * FF + i];
    __syncthreads();

    for (int p = t; p < PROJ; p += 256) {
        const int h = p / DD;
        float acc = 0.f;
        for (int f = 0; f < FF; ++f) acc += ctxs[h * FF + f] * Wv[(size_t)f * PROJ + p];
        attns[p] = acc;
    }
    __syncthreads();

    float acc = bo[t];
    for (int p = 0; p < PROJ; ++p) acc += attns[p] * Wo[(size_t)p * FF + t];
    out[(size_t)b * FF + t] = acc;
}

// ---------------------------------------------------------------------------
extern "C" void kernel_launch(void* const* d_in, const int* in_sizes, int n_in,
                              void* d_out, int out_size, void* d_ws, size_t ws_size,
                              hipStream_t stream) {
    const float* x  = (const float*)d_in[0];
    const float* Wq = (const float*)d_in[1];
    const float* Wk = (const float*)d_in[2];
    const float* Wv = (const float*)d_in[3];
    const float* Wo = (const float*)d_in[4];
    const float* bo = (const float*)d_in[5];
    float* out = (float*)d_out;

    float* ws   = (float*)d_ws;
    float* kq16 = ws;                                // B*16*F = 32768 floats
    float* w16  = kq16 + (size_t)BB * HP * FF;       // B*16*S = 262144 floats
    float* ctx  = w16 + (size_t)BB * HP * SS;        // B*H*F  = 16384 floats

    mhsa_kq_kernel<<<dim3(BB), dim3(256), 0, stream>>>(x, Wq, Wk, kq16);
    mhsa_score_kernel<<<dim3(BB, SS / 256), dim3(128), 0, stream>>>(x, kq16, w16);
    mhsa_softmax_kernel<<<dim3(BB * HH), dim3(256), 0, stream>>>(w16);
    mhsa_ctx_kernel<<<dim3(BB, FF / 64), dim3(128), 0, stream>>>(x, w16, ctx);
    mhsa_out_kernel<<<dim3(BB), dim3(256), 0, stream>>>(ctx, Wv, Wo, bo, out);
}